// MultiHeadAttention_66417374265491
// MI455X (gfx1250) — compile-verified
//
#include <hip/hip_runtime.h>
#include <math.h>

// Multi-head attention forward for MI455X (gfx1250, wave32, WMMA).
// B=2, S=2048, D=1024, H=16, HD=64.
// d_out = [ctx (B*S*D fp32)] ++ [attn (B*H*S*S fp32)].
//
// Roofline: attn tensor is 536 MB fp32; fusing scores+mask+softmax+ctx in one
// kernel writes it exactly once (~560 MB total HBM ~ 24 us @ 23.3 TB/s)
// instead of 2.1 GB for the unfused chain.  attn is streamed out with
// non-temporal stores so the write-once stream does not evict the L2-resident
// q/k/v working set (re-read ~128x per head).  fp32 WMMA keeps full precision
// at zero cost since the kernel is memory bound, not tensor-op bound.

#define B_  2
#define S_  2048
#define D_  1024
#define H_  16
#define HD_ 64
#define BH_ (B_ * H_)

// LDS score-strip row stride (floats): 2048 + 4 pad => stride mod 64 banks = 4,
// so column-wise v2f reads (lane r -> row r) hit banks {4r, 4r+1}: conflict-free.
#define LDS_STRIDE 2052

typedef __attribute__((ext_vector_type(2))) float v2f;
typedef __attribute__((ext_vector_type(4))) float v4f;
typedef __attribute__((ext_vector_type(8))) float v8f;

// D = A(16x4 f32) * B(4x16 f32) + C(16x16 f32), full fp32 precision.
static __device__ __forceinline__ v8f wmma_f32(v2f a, v2f b, v8f c) {
  return __builtin_amdgcn_wmma_f32_16x16x4_f32(false, a, false, b, (short)0, c,
                                               false, false);
}

// Fragment mapping (ISA 7.12.2, 32-bit):
//   A 16x4 : lane (hl*16+r) holds A[r][k0+2*hl+{0,1}] in {a.x,a.y}
//   B 4x16 : lane (hl*16+r) holds B[k0+2*hl+{0,1}][n0+r] in {b.x,b.y}
//   C 16x16: acc[e] = C[m0 + 8*hl + e][n0 + r]

// ---------------------------------------------------------------------------
// Kernel 1: fused QKV projection.  out = x @ W.T + b, written head-split as
// [BH, S, HD].  grid = (2, 256, 3), block = 256 (8 waves); wave -> 16x64 tile.
// ---------------------------------------------------------------------------
__global__ __launch_bounds__(256) void qkv_proj_kernel(
    const float* __restrict__ q_in, const float* __restrict__ k_in,
    const float* __restrict__ v_in, const float* __restrict__ Wq,
    const float* __restrict__ bq, const float* __restrict__ Wk,
    const float* __restrict__ bk, const float* __restrict__ Wv,
    const float* __restrict__ bv, float* __restrict__ qkv_ws) {
  const int mat = blockIdx.z;
  const float* x    = (mat == 0) ? q_in : (mat == 1) ? k_in : v_in;
  const float* W    = (mat == 0) ? Wq   : (mat == 1) ? Wk   : Wv;
  const float* bias = (mat == 0) ? bq   : (mat == 1) ? bk   : bv;
  float* out = qkv_ws + (size_t)mat * ((size_t)B_ * S_ * D_);

  const int wave = threadIdx.x >> 5;
  const int lane = threadIdx.x & 31;
  const int hl   = lane >> 4;   // half of the wave (K-group)
  const int r    = lane & 15;   // row/col within tile

  const int nBase = (blockIdx.x * 8 + wave) * 64;  // 64 output cols per wave
  const int mBase = blockIdx.y * 16;

  v8f acc[4] = {v8f{}, v8f{}, v8f{}, v8f{}};

  const float* aPtr = x + (size_t)(mBase + r) * D_ + 2 * hl;
  const float* bPtr = W + (size_t)(nBase + r) * D_ + 2 * hl;

  for (int k0 = 0; k0 < D_; k0 += 4) {
    v2f a = *(const v2f*)(aPtr + k0);
#pragma unroll
    for (int t = 0; t < 4; ++t) {
      v2f b = *(const v2f*)(bPtr + (size_t)t * 16 * D_ + k0);
      acc[t] = wmma_f32(a, b, acc[t]);
    }
  }

#pragma unroll
  for (int t = 0; t < 4; ++t) {
    const int n  = nBase + t * 16 + r;
    const float bb = bias[n];
    const int h  = n >> 6;    // n / HD
    const int hd = n & 63;    // n % HD
#pragma unroll
    for (int e = 0; e < 8; ++e) {
      const int m = mBase + hl * 8 + e;
      const int b = m >> 11;    // m / S
      const int s = m & 2047;   // m % S
      out[(((size_t)(b * H_ + h) * S_) + s) * HD_ + hd] = acc[t][e] + bb;
    }
  }
}

// ---------------------------------------------------------------------------
// Kernel 2: fused scores + mask + softmax + ctx.
// grid = (S/16 = 128 row-strips, BH = 32), block = 256 (8 waves).
// Workgroup owns a 16-row strip of one head's SxS score matrix, staged in a
// 16 x 2052 fp32 LDS buffer (128.25 KB -> 2 WGs per 320 KB WGP).
//   Stage 0: prefetch this wave's stage-3 v K-chunk (overlaps stages 1-2).
//   Stage 1: each wave -> scores for a 256-col range (WMMA), mask, -> LDS.
//   Stage 2: softmax in LDS (2 rows per wave); normalized attn -> d_out (NT).
//   Stage 3: ctx partial GEMM from LDS (wave = 256-wide K chunk), partials
//            stored back into the wave's own exhausted K columns of the strip.
//   Stage 4: reduce 8 partials, write ctx merged to [B,S,D].
// Mask quirk: tile(mask,(H,1,1)) => head-batch bh uses mask[bh % B].
// ---------------------------------------------------------------------------
__global__ __launch_bounds__(256) void attn_fused_kernel(
    const float* __restrict__ qh, const float* __restrict__ kh,
    const float* __restrict__ vh, const unsigned char* __restrict__ mask,
    float* __restrict__ attn_out, float* __restrict__ ctx_out) {
  __shared__ float sc[16 * LDS_STRIDE];  // 131,328 bytes

  const int bh    = blockIdx.y;
  const int mBase = blockIdx.x * 16;
  const int wave  = threadIdx.x >> 5;
  const int lane  = threadIdx.x & 31;
  const int hl    = lane >> 4;
  const int r     = lane & 15;
  const float ninf = -__builtin_inff();

  const float* vp = vh + (size_t)bh * S_ * HD_;

  // ---- Stage 0: prefetch this wave's v K-chunk (64 KB) for stage 3 --------
  // 512 cachelines of 128B; 16 per lane.  Lowers to global_prefetch_b8 and
  // overlaps the fetch with the score/softmax stages.
  {
    const float* vchunk = vp + (size_t)(wave * 256) * HD_;
#pragma unroll
    for (int i = 0; i < 16; ++i)
      __builtin_prefetch(vchunk + (size_t)(i * 32 + lane) * 32, 0, 3);
  }

  // ---- Stage 1: masked, scaled scores -> LDS ------------------------------
  {
    const float* qp = qh + ((size_t)bh * S_ + mBase + r) * HD_ + 2 * hl;
    // A fragments (q rows of this strip) are shared by all 4 column chunks:
    // load the 16 K-step fragments once.
    v2f afrag[16];
#pragma unroll
    for (int ks = 0; ks < 16; ++ks) afrag[ks] = *(const v2f*)(qp + 4 * ks);

    const unsigned char* mrow = mask + (size_t)(bh % B_) * S_ * S_;

#pragma unroll
    for (int c = 0; c < 4; ++c) {  // 4 chunks of 64 columns per wave
      const int nBase = wave * 256 + c * 64;
      const float* kp = kh + ((size_t)bh * S_ + nBase + r) * HD_ + 2 * hl;
      v8f acc[4] = {v8f{}, v8f{}, v8f{}, v8f{}};
#pragma unroll
      for (int ks = 0; ks < 16; ++ks) {
#pragma unroll
        for (int t = 0; t < 4; ++t) {
          v2f b = *(const v2f*)(kp + (size_t)t * 16 * HD_ + 4 * ks);
          acc[t] = wmma_f32(afrag[ks], b, acc[t]);
        }
      }
#pragma unroll
      for (int t = 0; t < 4; ++t) {
        const int n = nBase + t * 16 + r;
#pragma unroll
        for (int e = 0; e < 8; ++e) {
          const int m = hl * 8 + e;  // row within strip
          float s = acc[t][e] * 0.125f;  // 1/sqrt(64)
          if (mrow[(size_t)(mBase + m) * S_ + n]) s = ninf;
          sc[m * LDS_STRIDE + n] = s;
        }
      }
    }
  }
  __syncthreads();

  // ---- Stage 2: softmax rows in LDS; normalized attn -> global (NT) -------
  {
    float* astrip = attn_out + ((size_t)bh * S_ + mBase) * S_;
#pragma unroll
    for (int rr = 0; rr < 2; ++rr) {
      const int row = wave * 2 + rr;
      float* prow = sc + row * LDS_STRIDE;           // 16B-aligned (2052*4)
      const float4* p4 = (const float4*)prow;
      float4 vbuf[16];
      float mx = ninf;
#pragma unroll
      for (int i = 0; i < 16; ++i) {
        vbuf[i] = p4[i * 32 + lane];
        mx = fmaxf(mx, fmaxf(fmaxf(vbuf[i].x, vbuf[i].y),
                             fmaxf(vbuf[i].z, vbuf[i].w)));
      }
#pragma unroll
      for (int off = 16; off > 0; off >>= 1)
        mx = fmaxf(mx, __shfl_xor(mx, off, 32));

      float ssum = 0.f;
#pragma unroll
      for (int i = 0; i < 16; ++i) {
        vbuf[i].x = __expf(vbuf[i].x - mx);
        vbuf[i].y = __expf(vbuf[i].y - mx);
        vbuf[i].z = __expf(vbuf[i].z - mx);
        vbuf[i].w = __expf(vbuf[i].w - mx);
        ssum += (vbuf[i].x + vbuf[i].y) + (vbuf[i].z + vbuf[i].w);
      }
#pragma unroll
      for (int off = 16; off > 0; off >>= 1)
        ssum += __shfl_xor(ssum, off, 32);

      const float inv = 1.0f / ssum;
      v4f* g4 = (v4f*)(astrip + (size_t)row * S_);
      float4* l4 = (float4*)prow;
#pragma unroll
      for (int i = 0; i < 16; ++i) {
        float4 o;
        o.x = vbuf[i].x * inv;
        o.y = vbuf[i].y * inv;
        o.z = vbuf[i].z * inv;
        o.w = vbuf[i].w * inv;
        l4[i * 32 + lane] = o;  // normalized, for the ctx GEMM
        // attn output: write-once stream, keep it out of L2 (non-temporal).
        v4f onv = {o.x, o.y, o.z, o.w};
        __builtin_nontemporal_store(onv, &g4[i * 32 + lane]);
      }
    }
  }
  __syncthreads();

  // ---- Stage 3: ctx partial GEMM; wave = K chunk [wave*256, wave*256+256) --
  {
    const int kBase = wave * 256;
    v8f acc[4] = {v8f{}, v8f{}, v8f{}, v8f{}};
    for (int k0 = 0; k0 < 256; k0 += 4) {
      const int kk = kBase + k0 + 2 * hl;
      v2f a = *(const v2f*)(&sc[r * LDS_STRIDE + kk]);  // bank-conflict-free
#pragma unroll
      for (int t = 0; t < 4; ++t) {
        v2f b;
        b.x = vp[(size_t)kk * HD_ + t * 16 + r];
        b.y = vp[(size_t)(kk + 1) * HD_ + t * 16 + r];
        acc[t] = wmma_f32(a, b, acc[t]);
      }
    }
    // Store the 16x64 partial into this wave's own (fully consumed) K columns.
#pragma unroll
    for (int t = 0; t < 4; ++t) {
#pragma unroll
      for (int e = 0; e < 8; ++e) {
        const int m = hl * 8 + e;
        sc[m * LDS_STRIDE + kBase + t * 16 + r] = acc[t][e];
      }
    }
  }
  __syncthreads();

  // ---- Stage 4: reduce 8 partials, write ctx merged to [B,S,D] ------------
  {
    const int b = bh >> 4;   // bh / H
    const int h = bh & 15;   // bh % H
    for (int i = threadIdx.x; i < 16 * 64; i += 256) {
      const int m = i >> 6;
      const int n = i & 63;
      float s = 0.f;
#pragma unroll
      for (int w = 0; w < 8; ++w) s += sc[m * LDS_STRIDE + w * 256 + n];
      ctx_out[((size_t)b * S_ + mBase + m) * D_ + h * HD_ + n] = s;
    }
  }
}

// ---------------------------------------------------------------------------
extern "C" void kernel_launch(void* const* d_in, const int* in_sizes, int n_in,
                              void* d_out, int out_size, void* d_ws,
                              size_t ws_size, hipStream_t stream) {
  (void)in_sizes; (void)n_in; (void)out_size; (void)ws_size;

  const float* query = (const float*)d_in[0];
  const float* key_  = (const float*)d_in[1];
  const float* value = (const float*)d_in[2];
  const unsigned char* mask = (const unsigned char*)d_in[3];  // bool = 1 byte
  const float* Wq = (const float*)d_in[4];
  const float* bq = (const float*)d_in[5];
  const float* Wk = (const float*)d_in[6];
  const float* bk = (const float*)d_in[7];
  const float* Wv = (const float*)d_in[8];
  const float* bv = (const float*)d_in[9];

  float* ctx  = (float*)d_out;
  float* attn = ctx + (size_t)B_ * S_ * D_;  // 4,194,304 floats of ctx first

  float* ws = (float*)d_ws;
  const size_t headN = (size_t)B_ * S_ * D_;  // per-matrix head-split size
  float* qh = ws;
  float* kh = ws + headN;
  float* vh = ws + 2 * headN;

  // 1) QKV projections (fp32 WMMA): 3 GEMMs, head-split output.
  qkv_proj_kernel<<<dim3(2, 256, 3), 256, 0, stream>>>(
      query, key_, value, Wq, bq, Wk, bk, Wv, bv, ws);

  // 2) Fused scores + mask + softmax + ctx: attn written once, never re-read.
  attn_fused_kernel<<<dim3(S_ / 16, BH_), 256, 0, stream>>>(
      qh, kh, vh, mask, attn, ctx);
}